// HistogramObserver_29308856828601
// MI455X (gfx1250) — compile-verified
//
#include <hip/hip_runtime.h>
#include <hip/hip_bf16.h>

// -------------------- workspace layout (uint32 units) --------------------
// [0 .. 4095]   global coarse histogram (bits [30:19] of |x| pattern)
// [4096]        remaining rank (1-based, from the top)
// [4097]        selected key bits so far (high bits known, low bits zero)
// [4098]        order-encoded global min of x
// [4099]        candidate count
// [4100 .. ]    candidate keys (full 31-bit |x| patterns in the chosen bin)
#define HIST_BINS 4096
#define WS_RANK   4096
#define WS_KEY    4097
#define WS_MINKEY 4098
#define WS_CCOUNT 4099
#define WS_CAND   4100
#define CAND_CAP  65536u

__device__ __forceinline__ unsigned int umin32(unsigned int a, unsigned int b) {
    return a < b ? a : b;
}
// monotonic float -> u32 encoding (works for all non-NaN floats)
__device__ __forceinline__ unsigned int f2ord(float f) {
    unsigned int u = __float_as_uint(f);
    return (u & 0x80000000u) ? ~u : (u | 0x80000000u);
}
__device__ __forceinline__ float ord2f(unsigned int k) {
    unsigned int u = (k & 0x80000000u) ? (k & 0x7FFFFFFFu) : ~k;
    return __uint_as_float(u);
}

// -------------------- kernel 0: init workspace --------------------
__global__ void k_init(unsigned int* __restrict__ W, unsigned int m) {
    int i = blockIdx.x * blockDim.x + threadIdx.x;
    int stride = gridDim.x * blockDim.x;
    for (int j = i; j < HIST_BINS; j += stride) W[j] = 0u;
    if (i == 0) {
        W[WS_RANK]   = m;
        W[WS_KEY]    = 0u;
        W[WS_MINKEY] = 0xFFFFFFFFu;
        W[WS_CCOUNT] = 0u;
    }
}

// -------------------- kernel 1: coarse histogram + min (full read) ------
// Streams front-to-back so the *tail* of the array is L2-resident when the
// second pass starts (MI455X global L2 = 192 MB, input = 256 MB).
__global__ void __launch_bounds__(256)
k_pass1(const float* __restrict__ in, int n, unsigned int* __restrict__ W) {
    // 4-way wave-replicated LDS histogram: reduces same-address ds_add
    // contention across the 8 wave32s of the block. 64 KB of the 320 KB WGP LDS.
    __shared__ unsigned int lhist[4 * HIST_BINS];
    __shared__ unsigned int lmin;
    for (int b = threadIdx.x; b < 4 * HIST_BINS; b += blockDim.x) lhist[b] = 0u;
    if (threadIdx.x == 0) lmin = 0xFFFFFFFFu;
    __syncthreads();

    const unsigned int copy = ((threadIdx.x >> 5) & 3u) * HIST_BINS;
    const float4* __restrict__ in4 = (const float4*)in;
    const int n4 = n >> 2;
    const int tid = blockIdx.x * blockDim.x + threadIdx.x;
    const int stride = gridDim.x * blockDim.x;

    float mymin_f = __uint_as_float(0x7F800000u);  // +inf
    for (int i = tid; i < n4; i += stride) {
        __builtin_prefetch(in4 + i + stride, 0, 0);   // global_prefetch_b8
        float4 v = in4[i];                            // global_load_b128
        unsigned int a0 = __float_as_uint(v.x) & 0x7FFFFFFFu;
        unsigned int a1 = __float_as_uint(v.y) & 0x7FFFFFFFu;
        unsigned int a2 = __float_as_uint(v.z) & 0x7FFFFFFFu;
        unsigned int a3 = __float_as_uint(v.w) & 0x7FFFFFFFu;
        atomicAdd(&lhist[copy + (a0 >> 19)], 1u);     // ds_add_u32
        atomicAdd(&lhist[copy + (a1 >> 19)], 1u);
        atomicAdd(&lhist[copy + (a2 >> 19)], 1u);
        atomicAdd(&lhist[copy + (a3 >> 19)], 1u);
        mymin_f = fminf(mymin_f, fminf(fminf(v.x, v.y), fminf(v.z, v.w)));
    }
    for (int i = (n4 << 2) + tid; i < n; i += stride) {  // tail (n % 4)
        float v = in[i];
        unsigned int a = __float_as_uint(v) & 0x7FFFFFFFu;
        atomicAdd(&lhist[copy + (a >> 19)], 1u);
        mymin_f = fminf(mymin_f, v);
    }
    atomicMin(&lmin, f2ord(mymin_f));                 // ds_min_u32
    __syncthreads();

    for (int b = threadIdx.x; b < HIST_BINS; b += blockDim.x) {
        unsigned int c = lhist[b] + lhist[HIST_BINS + b] +
                         lhist[2 * HIST_BINS + b] + lhist[3 * HIST_BINS + b];
        if (c) atomicAdd(&W[b], c);
    }
    if (threadIdx.x == 0) atomicMin(&W[WS_MINKEY], lmin);
}

// -------------------- kernel 2: pick coarse bin (single block, 256 thr) --
__global__ void __launch_bounds__(256)
k_select_coarse(unsigned int* __restrict__ W) {
    __shared__ unsigned int part[256];
    __shared__ unsigned int suf[256];   // count strictly above each 16-bin chunk
    __shared__ unsigned int s_rank;
    const int t = threadIdx.x;

    unsigned int bins[16];
    unsigned int s = 0;
    #pragma unroll
    for (int j = 0; j < 16; ++j) { bins[j] = W[t * 16 + j]; s += bins[j]; }
    part[t] = s;
    if (t == 0) s_rank = W[WS_RANK];
    __syncthreads();
    if (t == 0) {
        unsigned int cum = 0;
        for (int c = 255; c >= 0; --c) { suf[c] = cum; cum += part[c]; }
    }
    __syncthreads();

    const unsigned int rank  = s_rank;
    const unsigned int above = suf[t];
    if (above < rank && above + part[t] >= rank) {     // exactly one thread
        unsigned int cum = above;
        #pragma unroll
        for (int j = 15; j >= 0; --j) {
            cum += bins[j];
            if (cum >= rank) {
                unsigned int bin = (unsigned int)(t * 16 + j);
                W[WS_KEY]  = bin << 19;
                W[WS_RANK] = rank - (cum - bins[j]);
                break;
            }
        }
    }
}

// -------------------- kernel 3: compact candidate bin (full read) -------
// Traverses in REVERSE order: pass 1 finished streaming front-to-back, so
// the last ~192 MB are L2-resident. Reading back-to-front serves ~75% of
// this pass from L2 instead of HBM; misses on the head region only evict
// lines already consumed. Lanes still touch consecutive addresses.
__global__ void __launch_bounds__(256)
k_pass2(const float* __restrict__ in, int n,
        unsigned int* __restrict__ W, unsigned int cap) {
    const unsigned int b1 = W[WS_KEY] >> 19;   // uniform -> scalar load
    const float4* __restrict__ in4 = (const float4*)in;
    const int n4 = n >> 2;
    const int tid = blockIdx.x * blockDim.x + threadIdx.x;
    const int stride = gridDim.x * blockDim.x;

    for (int i = tid; i < n4; i += stride) {
        const int j = n4 - 1 - i;                      // reverse traversal
        __builtin_prefetch(in4 + (j - stride), 0, 0);  // speculative: OK if OOB
        float4 v = in4[j];
        unsigned int a[4];
        a[0] = __float_as_uint(v.x) & 0x7FFFFFFFu;
        a[1] = __float_as_uint(v.y) & 0x7FFFFFFFu;
        a[2] = __float_as_uint(v.z) & 0x7FFFFFFFu;
        a[3] = __float_as_uint(v.w) & 0x7FFFFFFFu;
        #pragma unroll
        for (int jj = 0; jj < 4; ++jj) {
            if ((a[jj] >> 19) == b1) {
                unsigned int idx = atomicAdd(&W[WS_CCOUNT], 1u);
                if (idx < cap) W[WS_CAND + idx] = a[jj];
            }
        }
    }
    for (int i = (n4 << 2) + tid; i < n; i += stride) {  // tail (n % 4)
        unsigned int a = __float_as_uint(in[i]) & 0x7FFFFFFFu;
        if ((a >> 19) == b1) {
            unsigned int idx = atomicAdd(&W[WS_CCOUNT], 1u);
            if (idx < cap) W[WS_CAND + idx] = a;
        }
    }
}

// -------------------- kernel 4: exact select among candidates -----------
// Refine low 19 bits in 8/8/3-bit rounds via LDS histograms.
__global__ void __launch_bounds__(512)
k_select_fine(unsigned int* __restrict__ W, unsigned int cap) {
    __shared__ unsigned int lhist[256];
    __shared__ unsigned int s_key, s_rank, s_cnt;
    const int t = threadIdx.x;
    if (t == 0) {
        s_key  = W[WS_KEY];
        s_rank = W[WS_RANK];
        s_cnt  = umin32(W[WS_CCOUNT], cap);
    }
    __syncthreads();

    const int shifts[3] = {11, 3, 0};
    const int widths[3] = {8, 8, 3};
    for (int r = 0; r < 3; ++r) {
        const int sft = shifts[r], w = widths[r];
        const unsigned int nb = 1u << w;
        for (int b = t; b < 256; b += blockDim.x) lhist[b] = 0u;
        __syncthreads();
        const unsigned int keyHi = s_key >> (sft + w);
        const unsigned int cnt = s_cnt;
        for (unsigned int i = (unsigned int)t; i < cnt; i += blockDim.x) {
            unsigned int k = W[WS_CAND + i];
            if ((k >> (sft + w)) == keyHi)
                atomicAdd(&lhist[(k >> sft) & (nb - 1u)], 1u);
        }
        __syncthreads();
        if (t == 0) {
            unsigned int rank = s_rank, cum = 0;
            for (int b = (int)nb - 1; b >= 0; --b) {
                cum += lhist[b];
                if (cum >= rank) {
                    s_key |= ((unsigned int)b) << sft;
                    s_rank = rank - (cum - lhist[b]);
                    break;
                }
            }
        }
        __syncthreads();
    }
    if (t == 0) W[WS_KEY] = s_key;   // exact |x| bit pattern of m-th largest
}

// -------------------- kernel 5: momentum update + write output ----------
__global__ void k_final(const unsigned int* __restrict__ W,
                        const float* __restrict__ min_val,
                        const float* __restrict__ max_val,
                        const int* __restrict__ num_flag,
                        float* __restrict__ out) {
    float max_cur = __uint_as_float(W[WS_KEY]);   // sign bit is 0: value itself
    float min_cur = ord2f(W[WS_MINKEY]);
    if (*num_flag == 0) {
        out[0] = max_cur;
        out[1] = min_cur;
    } else {
        out[0] = 0.9f * max_val[0] + 0.1f * max_cur;
        out[1] = 0.9f * min_val[0] + 0.1f * min_cur;
    }
}

// -------------------- launch --------------------
extern "C" void kernel_launch(void* const* d_in, const int* in_sizes, int n_in,
                              void* d_out, int out_size, void* d_ws, size_t ws_size,
                              hipStream_t stream) {
    (void)n_in; (void)out_size;
    const float* in       = (const float*)d_in[0];
    const float* min_val  = (const float*)d_in[1];
    const float* max_val  = (const float*)d_in[2];
    const int*   num_flag = (const int*)d_in[3];
    float* out = (float*)d_out;
    unsigned int* W = (unsigned int*)d_ws;

    const long long n = (long long)in_sizes[0];
    // Match Python: k = int(0.9999 * n) (double multiply, truncate toward zero)
    long long k = (long long)(0.9999 * (double)n);
    if (k < 1) k = 1;
    const unsigned int m = (unsigned int)(n - k + 1);

    // candidate capacity from available workspace
    unsigned int cap = 0;
    size_t words = ws_size / 4;
    if (words > (size_t)WS_CAND) {
        size_t c = words - (size_t)WS_CAND;
        cap = (unsigned int)(c > (size_t)CAND_CAP ? CAND_CAP : c);
    }

    const int threads = 256;
    long long n4 = n >> 2;
    long long want = (n4 + threads * 8 - 1) / (threads * 8);
    int blocks = (int)(want < 1 ? 1 : (want > 4096 ? 4096 : want));

    k_init<<<16, 256, 0, stream>>>(W, m);
    k_pass1<<<blocks, threads, 0, stream>>>(in, (int)n, W);
    k_select_coarse<<<1, 256, 0, stream>>>(W);
    k_pass2<<<blocks, threads, 0, stream>>>(in, (int)n, W, cap);
    k_select_fine<<<1, 512, 0, stream>>>(W, cap);
    k_final<<<1, 1, 0, stream>>>(W, min_val, max_val, num_flag, out);
}